// Net_52209622450626
// MI455X (gfx1250) — compile-verified
//
#include <hip/hip_runtime.h>
#include <math.h>

typedef __attribute__((ext_vector_type(2))) float v2f;
typedef __attribute__((ext_vector_type(8))) float v8f;

#define Tn 256
#define Kn 32
#define Vn 32
#define ST 36              // padded LDS row stride in floats (144B, 16B-aligned rows)
#define NSTEP (Tn + Kn)    // 288 scan steps

__device__ __forceinline__ float myrelu(float z) {
    return z < 0.f ? 0.01f * z : (z > 1.f ? fmaf(0.01f, z, 0.99f) : z);
}

struct BF { v2f b[2][8]; };   // B fragments: 2 N-tiles x 8 k-steps

// B layout (16x16x4 f32 WMMA, 4x16 B): VGPR j holds row K=j (lanes 0-15) and
// K=j+2 (lanes 16-31), N = lane&15 (+16 per N-tile).
__device__ __forceinline__ void load_bfrags(const float* __restrict__ bm, int lane, BF& bf) {
    const int half = lane >> 4;
    const int n0 = lane & 15;
#pragma unroll
    for (int nt = 0; nt < 2; ++nt) {
#pragma unroll
        for (int s = 0; s < 8; ++s) {
            const int kk = 4 * s + 2 * half;
            const int n = n0 + 16 * nt;
            v2f bb;
            bb.x = bm[kk * Kn + n];
            bb.y = bm[(kk + 1) * Kn + n];
            bf.b[nt][s] = bb;
        }
    }
}

// One wave computes 2 M-tiles x 2 N-tiles of (256x32)@(32x32) via f32 WMMA.
// A layout: lane<16 holds A[16m+lane][4s+0..1], lane>=16 holds A[16m+lane-16][4s+2..3].
// C/D layout: VGPR i at lane l = C[16m + i + 8*(l>>4)][(l&15) + 16*nt].
__device__ __forceinline__ void gemm_tiles(const float* __restrict__ Asrc, int astride,
                                           float* __restrict__ Rdst, const BF& bf,
                                           int wave, int lane) {
    const int half = lane >> 4;
    const int l15 = lane & 15;
#pragma unroll
    for (int mt = 0; mt < 2; ++mt) {
        const int m = 2 * wave + mt;
        const float* arow = Asrc + (16 * m + l15) * astride + 2 * half;
        v8f acc0 = {};
        v8f acc1 = {};
#pragma unroll
        for (int s = 0; s < 8; ++s) {
            v2f a = *(const v2f*)(arow + 4 * s);
            acc0 = __builtin_amdgcn_wmma_f32_16x16x4_f32(false, a, false, bf.b[0][s],
                                                         (short)0, acc0, false, false);
            acc1 = __builtin_amdgcn_wmma_f32_16x16x4_f32(false, a, false, bf.b[1][s],
                                                         (short)0, acc1, false, false);
        }
        const int rbase = (16 * m + 8 * half) * ST + l15;
#pragma unroll
        for (int i = 0; i < 8; ++i) {
            Rdst[rbase + i * ST] = acc0[i];
            Rdst[rbase + i * ST + 16] = acc1[i];
        }
    }
}

__global__ __launch_bounds__(256) void scan_kernel(const float* __restrict__ x,
                                                   const float* __restrict__ w_right,
                                                   const float* __restrict__ w_op,
                                                   float* __restrict__ out) {
    __shared__ float sX[2][Tn * ST];   // ping-pong state / "right" buffers
    __shared__ float sRW[Kn * Kn];     // right_w as B matrix [k][j]
    __shared__ float sAW[Vn * Kn];     // atom_w as B matrix [v][k]
    __shared__ float4 sPK[Kn];         // per-k packed constants {op1-op2, op3, op2+op4, op4}

    const int tid = threadIdx.x;
    const int lane = tid & 31;
    const int wave = tid >> 5;
    const int b = blockIdx.x;
    const int t = tid;                 // elementwise row owned by this thread

    // ---- setup: tiny softmaxes (one row per thread, first 32 threads) ----
    if (tid < Kn) {
        const int j = tid;
        {   // softmax over w_right row j (33 entries), keep first 32, transposed
            const float* row = w_right + j * (Kn + 1);
            float mx = row[0];
            for (int i = 1; i < Kn + 1; ++i) mx = fmaxf(mx, row[i]);
            float sum = 0.f;
            for (int i = 0; i < Kn + 1; ++i) sum += __expf(row[i] - mx);
            const float inv = 1.f / sum;
            for (int k = 0; k < Kn; ++k) sRW[k * Kn + j] = __expf(row[k] - mx) * inv;
        }
        {   // softmax over w_op row j (37 entries)
            const float* row = w_op + j * (5 + Vn);
            float mx = row[0];
            for (int i = 1; i < 5 + Vn; ++i) mx = fmaxf(mx, row[i]);
            float sum = 0.f;
            for (int i = 0; i < 5 + Vn; ++i) sum += __expf(row[i] - mx);
            const float inv = 1.f / sum;
            const float op1 = __expf(row[1] - mx) * inv;
            const float op2 = __expf(row[2] - mx) * inv;
            const float op3 = __expf(row[3] - mx) * inv;
            const float op4 = __expf(row[4] - mx) * inv;
            sPK[j] = make_float4(op1 - op2, op3, op2 + op4, op4);
            for (int v = 0; v < Vn; ++v) sAW[v * Kn + j] = __expf(row[5 + v] - mx) * inv;
        }
    }
    __syncthreads();

    // ---- atom_x = x[b] @ atom_w via WMMA, staged through sX[1] ----
    BF bf;
    load_bfrags(sAW, lane, bf);
    gemm_tiles(x + (size_t)b * Tn * Vn, Vn, sX[1], bf, wave, lane);
    __syncthreads();

    // Pull my atom row into registers, fold the (op1-op2) bias in; zero sX[0].
    float atomreg[Kn];
#pragma unroll
    for (int c = 0; c < 8; ++c) {
        float4 q = *(const float4*)&sX[1][t * ST + 4 * c];
        atomreg[4 * c + 0] = q.x; atomreg[4 * c + 1] = q.y;
        atomreg[4 * c + 2] = q.z; atomreg[4 * c + 3] = q.w;
    }
#pragma unroll
    for (int k = 0; k < Kn; ++k) atomreg[k] += sPK[k].x;

    load_bfrags(sRW, lane, bf);        // loop-invariant right_w B fragments (persist)
    {
        const float4 z = make_float4(0.f, 0.f, 0.f, 0.f);
#pragma unroll
        for (int c = 0; c < 8; ++c) *(float4*)&sX[0][t * ST + 4 * c] = z;
    }
    __syncthreads();

    // ---- 288-step scan, fully LDS-resident ----
    int cur = 0;
    for (int step = 0; step < NSTEP; ++step) {
        float* Xc = sX[cur];
        float* Rb = sX[cur ^ 1];
        gemm_tiles(Xc, ST, Rb, bf, wave, lane);   // Rb = Xc @ right_w
        __syncthreads();

        // Elementwise: thread t owns row t; new row overwrites Rb row t.
        float xr[Kn + 1], xn[Kn + 1];
#pragma unroll
        for (int c = 0; c < 8; ++c) {
            float4 q = *(const float4*)&Xc[t * ST + 4 * c];
            xr[4 * c] = q.x; xr[4 * c + 1] = q.y; xr[4 * c + 2] = q.z; xr[4 * c + 3] = q.w;
        }
        xr[Kn] = 0.f;
        if (t < Tn - 1) {
#pragma unroll
            for (int c = 0; c < 8; ++c) {
                float4 q = *(const float4*)&Xc[(t + 1) * ST + 4 * c];
                xn[4 * c] = q.x; xn[4 * c + 1] = q.y; xn[4 * c + 2] = q.z; xn[4 * c + 3] = q.w;
            }
        } else {
#pragma unroll
            for (int k = 0; k < Kn; ++k) xn[k] = 0.f;
        }
        xn[Kn] = 0.f;

#pragma unroll
        for (int c = 0; c < 8; ++c) {
            float4 rv = *(const float4*)&Rb[t * ST + 4 * c];
            const float rj[4] = {rv.x, rv.y, rv.z, rv.w};
            float nj[4];
#pragma unroll
            for (int j = 0; j < 4; ++j) {
                const int k = 4 * c + j;
                const float4 pk = sPK[k];                       // LDS broadcast
                const float left = xr[k + 1];
                const float u2 = myrelu(left + xn[k] - 1.f);
                const float v = atomreg[k] - left * pk.x + xn[k + 1] * pk.y
                              + rj[j] * pk.z + u2 * pk.w;
                nj[j] = myrelu(v);
            }
            *(float4*)&Rb[t * ST + 4 * c] = make_float4(nj[0], nj[1], nj[2], nj[3]);
        }
        __syncthreads();
        cur ^= 1;
    }

    // ---- epilogue: sigmoid((X - 0.5) * 5) ----
    const float* Xf = sX[cur];
    float* og = out + ((size_t)b * Tn + t) * Kn;
#pragma unroll
    for (int c = 0; c < 8; ++c) {
        float4 q = *(const float4*)&Xf[t * ST + 4 * c];
        float4 o;
        o.x = 1.f / (1.f + __expf(-5.f * (q.x - 0.5f)));
        o.y = 1.f / (1.f + __expf(-5.f * (q.y - 0.5f)));
        o.z = 1.f / (1.f + __expf(-5.f * (q.z - 0.5f)));
        o.w = 1.f / (1.f + __expf(-5.f * (q.w - 0.5f)));
        *(float4*)&og[4 * c] = o;
    }
}

extern "C" void kernel_launch(void* const* d_in, const int* in_sizes, int n_in,
                              void* d_out, int out_size, void* d_ws, size_t ws_size,
                              hipStream_t stream) {
    const float* x       = (const float*)d_in[0];   // (64, 256, 32)
    const float* w_right = (const float*)d_in[1];   // (32, 33)
    const float* w_op    = (const float*)d_in[2];   // (32, 37)
    float* out = (float*)d_out;                     // (64, 256, 32)
    (void)in_sizes; (void)n_in; (void)out_size; (void)d_ws; (void)ws_size;

    scan_kernel<<<64, 256, 0, stream>>>(x, w_right, w_op, out);
}